// MetaKernel_86285892976671
// MI455X (gfx1250) — compile-verified
//
#include <hip/hip_runtime.h>

typedef __attribute__((ext_vector_type(2))) float v2f;
typedef __attribute__((ext_vector_type(8))) float v8f;

#define B_  2
#define CF_ 64
#define H_  64
#define W_  2048
#define NKK 9            // 3x3 kernel positions
#define TP  16           // pixels per block (one WMMA N-tile)
#define NTHREADS 288     // 9 waves of 32, one wave per kernel position

// ---------------------------------------------------------------------------
// Kernel 0: fuse the two linear layers: wc[o][c] = sum_k w1[o,k] * w0[k,c].
// Stored padded to K=4 (wc[o*4+3] = 0) so it drops straight into WMMA A.
// ---------------------------------------------------------------------------
__global__ void fuse_weights_kernel(const float* __restrict__ w0,   // [16,3]
                                    const float* __restrict__ w1,   // [64,16]
                                    float* __restrict__ wc)         // [64,4]
{
    int o = threadIdx.x;
    if (o < 64) {
        float a0 = 0.f, a1 = 0.f, a2 = 0.f;
        #pragma unroll
        for (int k = 0; k < 16; ++k) {
            float w = w1[o * 16 + k];
            a0 += w * w0[k * 3 + 0];
            a1 += w * w0[k * 3 + 1];
            a2 += w * w0[k * 3 + 2];
        }
        wc[o * 4 + 0] = a0;
        wc[o * 4 + 1] = a1;
        wc[o * 4 + 2] = a2;
        wc[o * 4 + 3] = 0.f;
    }
}

// ---------------------------------------------------------------------------
// Main kernel: one block = 16 consecutive pixels of one (b, h) row.
// Phase 1: wave n (0..8) builds rel[3][16] for its kernel offset and runs
//          4x V_WMMA_F32_16X16X4_F32 (M=64, K=4, N=16) against the fused
//          weight matrix; ReLU'd results go to LDS in [c*9+n][p] order.
// Phase 2: all threads stream output: b128 LDS weight reads, feature loads,
//          b128 global stores.
// ---------------------------------------------------------------------------
__global__ __launch_bounds__(NTHREADS) void pacnet_weights_kernel(
    const float* __restrict__ features,  // [B, 64, H, W]
    const float* __restrict__ coord,     // [B, 3, H, W]
    const float* __restrict__ wc,        // [64, 4] fused (K padded)
    float* __restrict__ out)             // [B, 576, H, W]
{
    __shared__ float wlds[CF_ * NKK * TP];   // 36 KB: [cn = c*9+n][p]

    const int tid  = threadIdx.x;
    const int lane = tid & 31;
    const int wave = tid >> 5;               // n = kernel position 0..8
    const int blk  = blockIdx.x;
    const int wt   = blk & 127;              // W/16 = 128 tiles per row
    const int h    = (blk >> 7) & 63;
    const int b    = blk >> 13;
    const int w0p  = wt * TP;

    // ---------------- phase 1: weights via WMMA ----------------
    const int n  = wave;
    const int di = n / 3 - 1;
    const int dj = n % 3 - 1;
    const int p  = lane & 15;                // column (pixel) / A-matrix row
    const int hi = lane >> 4;                // 0: K rows {0,1}, 1: K rows {2,3}

    const int  hh   = h + di;
    const int  ww   = w0p + p + dj;
    const bool nbin = (hh >= 0) && (hh < H_) && (ww >= 0) && (ww < W_);

    // B matrix (4x16 rel): lanes 0-15 -> rel[0],rel[1]; lanes 16-31 -> rel[2],0
    v2f bm;
    {
        const int c0 = hi ? 2 : 0;
        const float* cb0 = coord + ((size_t)b * 3 + c0) * (size_t)(H_ * W_);
        float cen0 = cb0[h * W_ + (w0p + p)];
        float nb0  = nbin ? cb0[hh * W_ + ww] : 0.f;
        float r0   = (cen0 == -1.0f) ? 0.f : (nb0 - cen0);
        float r1   = 0.f;
        if (!hi) {
            const float* cb1 = coord + ((size_t)b * 3 + 1) * (size_t)(H_ * W_);
            float cen1 = cb1[h * W_ + (w0p + p)];
            float nb1  = nbin ? cb1[hh * W_ + ww] : 0.f;
            r1 = (cen1 == -1.0f) ? 0.f : (nb1 - cen1);
        }
        bm.x = r0;
        bm.y = r1;
    }

    // 4 M-tiles of the 64x4 fused weight matrix
    #pragma unroll
    for (int m = 0; m < 4; ++m) {
        const int o = m * 16 + p;            // A-matrix row (output channel)
        v2f am;
        am.x = wc[o * 4 + 2 * hi + 0];
        am.y = wc[o * 4 + 2 * hi + 1];

        v8f acc = {};
        acc = __builtin_amdgcn_wmma_f32_16x16x4_f32(
            /*neg_a=*/false, am, /*neg_b=*/false, bm,
            /*c_mod=*/(short)0, acc, /*reuse_a=*/false, /*reuse_b=*/false);

        // D layout: lane<16 -> M=v, N=lane ; lane>=16 -> M=v+8, N=lane-16
        #pragma unroll
        for (int v = 0; v < 8; ++v) {
            const int oo = m * 16 + v + hi * 8;   // output channel c
            const int cn = oo * NKK + n;          // c*9 + n : output order
            wlds[cn * TP + p] = fmaxf(acc[v], 0.f);
        }
    }
    __syncthreads();

    // ---------------- phase 2: elementwise multiply + store ----------------
    // 64*9*16 floats = 2304 float4 groups; 2304 / 288 = 8 iterations.
    const float* fb = features + (size_t)b * CF_ * H_ * W_;
    float*       ob = out      + (size_t)b * (CF_ * NKK) * H_ * W_;

    #pragma unroll
    for (int it = 0; it < 8; ++it) {
        const int idx = tid + it * NTHREADS;
        const int pq  = idx & 3;             // which float4 within the 16 pixels
        const int cn  = idx >> 2;            // c*9 + n
        const int c   = cn / 9;
        const int n2  = cn - c * 9;
        const int d2i = n2 / 3 - 1;
        const int d2j = n2 - (n2 / 3) * 3 - 1;
        const int hh2 = h + d2i;
        const int wb  = w0p + pq * 4;

        const float4 wv = *(const float4*)&wlds[cn * TP + pq * 4];

        float f0 = 0.f, f1 = 0.f, f2 = 0.f, f3 = 0.f;
        if (hh2 >= 0 && hh2 < H_) {
            const float* frow = fb + ((size_t)c * H_ + hh2) * W_;
            const int ws = wb + d2j;
            // scalar loads: dj=+-1 makes b128 misaligned; features are
            // L2-resident so per-dword loads are cheap.
            if (ws + 0 >= 0 && ws + 0 < W_) f0 = frow[ws + 0];
            if (ws + 1 >= 0 && ws + 1 < W_) f1 = frow[ws + 1];
            if (ws + 2 >= 0 && ws + 2 < W_) f2 = frow[ws + 2];
            if (ws + 3 >= 0 && ws + 3 < W_) f3 = frow[ws + 3];
        }

        float4 ov;
        ov.x = f0 * wv.x;
        ov.y = f1 * wv.y;
        ov.z = f2 * wv.z;
        ov.w = f3 * wv.w;
        *(float4*)(ob + ((size_t)cn * H_ + h) * W_ + wb) = ov;   // b128 store
    }
}

// ---------------------------------------------------------------------------
extern "C" void kernel_launch(void* const* d_in, const int* in_sizes, int n_in,
                              void* d_out, int out_size, void* d_ws, size_t ws_size,
                              hipStream_t stream) {
    const float* features = (const float*)d_in[0];  // [2,64,64,2048]
    const float* coord    = (const float*)d_in[1];  // [2,3,64,2048]
    const float* w0       = (const float*)d_in[2];  // [16,3]
    const float* w1       = (const float*)d_in[3];  // [64,16]
    float*       out      = (float*)d_out;          // [2,576,64,2048]
    float*       wc       = (float*)d_ws;           // [64,4] fused weights

    fuse_weights_kernel<<<1, 64, 0, stream>>>(w0, w1, wc);

    const int nblocks = B_ * H_ * (W_ / TP);        // 2*64*128 = 16384
    pacnet_weights_kernel<<<nblocks, NTHREADS, 0, stream>>>(features, coord, wc, out);
}